// LSH_27247272526202
// MI455X (gfx1250) — compile-verified
//
#include <hip/hip_runtime.h>
#include <math.h>

// ---------------- problem constants ----------------
#define B_    8
#define S_    8192
#define D_    128
#define NH_   8
#define NB_   128     // buckets per hash
#define NHALF_ 64     // nb/2 (rotation cols per hash)
#define NCH_  1024    // number of chunks (= nh*nb)
#define M_    64      // queries per chunk
#define NKEY_ 128     // keys per chunk (chunk + look-back)
#define NPOSBLK_ 32   // sort blocks per (b,h): 8192/256

typedef __attribute__((ext_vector_type(2))) float v2f;
typedef __attribute__((ext_vector_type(8))) float v8f;
typedef __attribute__((ext_vector_type(4))) unsigned int u32x4;
typedef __attribute__((ext_vector_type(4))) int i32x4;
typedef __attribute__((ext_vector_type(8))) int i32x8;

#define NEG_FLT_MAX (-3.402823466e38f)
#define SELF_ATTN   (-50000.0f)
#define SCALE_      (0.08838834764831845f)   // 128^-0.5

__device__ __forceinline__ v8f wmma_f32_k4(v8f acc, v2f a, v2f b) {
    // D = A(16x4,f32) * B(4x16,f32) + C(16x16,f32)
    return __builtin_amdgcn_wmma_f32_16x16x4_f32(
        /*neg_a=*/false, a, /*neg_b=*/false, b,
        /*c_mod=*/(short)0, acc, /*reuse_a=*/false, /*reuse_b=*/false);
}

// ---------------------------------------------------------------------
// Tensor Data Mover helpers (D# descriptor per CDNA5 ISA)
// ---------------------------------------------------------------------
__device__ __forceinline__ void tdm_issue(u32x4 g0, i32x8 g1, i32x4 g2, i32x4 g3) {
#if __clang_major__ >= 23
    i32x8 z8 = {0, 0, 0, 0, 0, 0, 0, 0};
    __builtin_amdgcn_tensor_load_to_lds(g0, g1, g2, g3, z8, 0);
#else
    __builtin_amdgcn_tensor_load_to_lds(g0, g1, g2, g3, 0);
#endif
}

// group0: [1:0]=count(1), [30]=gather_index_size, [31]=gather_mode,
//         [63:32]=lds_addr, [120:64]=global_addr, [127:126]=type(2)
__device__ __forceinline__ u32x4 tdm_g0(unsigned lds_addr, const void* gptr,
                                        unsigned flags) {
    const unsigned long long ga = (unsigned long long)(__UINTPTR_TYPE__)gptr;
    u32x4 g0;
    g0.x = 1u | flags;
    g0.y = lds_addr;
    g0.z = (unsigned)ga;
    g0.w = (unsigned)((ga >> 32) & 0x01ffffffu) | (2u << 30);
    return g0;
}

// group1: data_size=2 (4B), tensor_dim0[79:48], tensor_dim1[111:80],
//         tile_dim0[127:112], tile_dim1[143:128], tensor_dim0_stride[207:160]
__device__ __forceinline__ i32x8 tdm_g1(unsigned tensor_d0, unsigned tensor_d1,
                                        unsigned tile_d0, unsigned tile_d1,
                                        unsigned long long stride0) {
    const unsigned long long q0 = ((unsigned long long)(2u << 16))
                                | ((unsigned long long)(tensor_d0 & 0xffffu) << 48);
    const unsigned long long q1 = ((unsigned long long)((tensor_d0 >> 16) & 0xffffu))
                                | ((unsigned long long)tensor_d1 << 16)
                                | ((unsigned long long)(tile_d0 & 0xffffu) << 48);
    const unsigned long long q2 = ((unsigned long long)(tile_d1 & 0xffffu))
                                | ((stride0 & 0xffffffffull) << 32);
    const unsigned long long q3 = (stride0 >> 32) & 0xffffull;
    i32x8 g1;
    g1[0] = (int)(unsigned)q0; g1[1] = (int)(unsigned)(q0 >> 32);
    g1[2] = (int)(unsigned)q1; g1[3] = (int)(unsigned)(q1 >> 32);
    g1[4] = (int)(unsigned)q2; g1[5] = (int)(unsigned)(q2 >> 32);
    g1[6] = (int)(unsigned)q3; g1[7] = (int)(unsigned)(q3 >> 32);
    return g1;
}

// pack 8 row indices (16-bit each) into one descriptor index group
__device__ __forceinline__ i32x4 tdm_pack_idx8(const int* p) {
    i32x4 g;
    g[0] = (p[0] & 0xffff) | (p[1] << 16);
    g[1] = (p[2] & 0xffff) | (p[3] << 16);
    g[2] = (p[4] & 0xffff) | (p[5] << 16);
    g[3] = (p[6] & 0xffff) | (p[7] << 16);
    return g;
}

// =====================================================================
// k1: rotated = vecs @ rot  (per hash slice), argmax([r,-r]) -> bucket
// grid: (128 row-blocks, NH, B) x 256 threads
// dyn LDS: Qs 64x128 | Ws 128x64 (k-major) | Rs 64x64  = 80 KB
// Staging via TDM: 1D 32KB descriptor for Q, 2D 128x64 tile for W_h.
// =====================================================================
__global__ __launch_bounds__(256)
void k1_buckets(const float* __restrict__ vecs, const float* __restrict__ rot,
                int* __restrict__ buckets) {
    extern __shared__ float smem[];
    float* Qs = smem;                  // [64][128]
    float* Ws = Qs + 64 * 128;         // [128][64]
    float* Rs = Ws + 128 * 64;         // [64][64]

    const int rb = blockIdx.x;         // row block (64 rows)
    const int h  = blockIdx.y;
    const int b  = blockIdx.z;
    const int t  = threadIdx.x;
    const int lane = t & 31;
    const int wave = t >> 5;

    if (wave == 0) {
        const i32x4 z4 = {0, 0, 0, 0};
        const unsigned qs_off = (unsigned)(__UINTPTR_TYPE__)(void*)Qs;
        const unsigned ws_off = (unsigned)(__UINTPTR_TYPE__)(void*)Ws;
        // Q: contiguous 64x128 f32 block as a 1D tile
        tdm_issue(tdm_g0(qs_off, vecs + ((size_t)b * S_ + (size_t)rb * 64) * D_, 0u),
                  tdm_g1(/*t0=*/64 * D_, /*t1=*/1, /*tile0=*/64 * D_, /*tile1=*/0,
                         /*stride0=*/64 * D_), z4, z4);
        // W_h: 128 rows x 64 cols tile out of the 128x512 rotation tensor
        tdm_issue(tdm_g0(ws_off, rot + (size_t)h * NHALF_, 0u),
                  tdm_g1(/*t0=*/NH_ * NHALF_, /*t1=*/D_, /*tile0=*/NHALF_,
                         /*tile1=*/D_, /*stride0=*/NH_ * NHALF_), z4, z4);
        __builtin_amdgcn_s_wait_tensorcnt(0);
    }
    __syncthreads();

    const int lrow = lane & 15;
    const int hi   = lane >> 4;        // 0: K={0,1}/M rows 0-7 ; 1: K={2,3}/M rows 8-15
    #pragma unroll
    for (int q = 0; q < 2; ++q) {
        const int tile = wave * 2 + q;          // 16 tiles: 4 Mt x 4 Nt
        const int mt = tile >> 2, nt = tile & 3;
        v8f acc = {0.f,0.f,0.f,0.f,0.f,0.f,0.f,0.f};
        const float* qrow = Qs + (mt * 16 + lrow) * 128;
        for (int kk = 0; kk < 32; ++kk) {
            const int k0 = kk * 4 + hi * 2;
            v2f a = *(const v2f*)(qrow + k0);                       // A frag
            v2f bb;
            bb.x = Ws[(k0    ) * 64 + nt * 16 + lrow];              // B frag (k-major)
            bb.y = Ws[(k0 + 1) * 64 + nt * 16 + lrow];
            acc = wmma_f32_k4(acc, a, bb);
        }
        #pragma unroll
        for (int r = 0; r < 8; ++r) {
            const int row = mt * 16 + hi * 8 + r;
            Rs[row * 64 + nt * 16 + lrow] = acc[r];
        }
    }
    __syncthreads();

    // per-row argmax over concat([r, -r]) with jnp first-occurrence ties
    if (t < 64) {
        float maxv = NEG_FLT_MAX, minv = 3.402823466e38f;
        int imax = 0, imin = 0;
        const float* row = Rs + t * 64;
        for (int i = 0; i < 64; ++i) {
            const float v = row[i];
            if (v > maxv) { maxv = v; imax = i; }
            if (v < minv) { minv = v; imin = i; }
        }
        const int bucket = (maxv >= -minv) ? imax : (64 + imin);
        const int pos = rb * 64 + t;
        buckets[(size_t)b * (NH_ * S_) + (size_t)h * S_ + pos] = h * NB_ + bucket;
    }
}

// =====================================================================
// k2: per-(b,h,posblock) 128-bin histogram (LDS, no global init needed)
// =====================================================================
__global__ __launch_bounds__(256)
void k2_hist(const int* __restrict__ buckets, int* __restrict__ blockhist) {
    __shared__ int hist[NB_];
    const int blk = blockIdx.x, h = blockIdx.y, b = blockIdx.z;
    const int t = threadIdx.x;
    if (t < NB_) hist[t] = 0;
    __syncthreads();
    const int pos = blk * 256 + t;
    const int bin = buckets[(size_t)b * (NH_ * S_) + (size_t)h * S_ + pos] & (NB_ - 1);
    atomicAdd(&hist[bin], 1);
    __syncthreads();
    if (t < NB_)
        blockhist[((((size_t)b * NH_ + h) * NPOSBLK_) + blk) * NB_ + t] = hist[t];
}

// =====================================================================
// k3a: exclusive scan over pos-blocks per (b,h,bin); emit bin totals
// =====================================================================
__global__ __launch_bounds__(256)
void k3a_scanblocks(int* __restrict__ blockhist, int* __restrict__ bin_total) {
    const int idx = blockIdx.x * 256 + threadIdx.x;   // 0..8191
    if (idx >= B_ * NH_ * NB_) return;
    const int bin = idx & (NB_ - 1);
    const int h   = (idx >> 7) & 7;
    const int b   = idx >> 10;
    const size_t base = (((size_t)b * NH_ + h) * NPOSBLK_) * NB_ + bin;
    int run = 0;
    for (int blk = 0; blk < NPOSBLK_; ++blk) {
        const int c = blockhist[base + (size_t)blk * NB_];
        blockhist[base + (size_t)blk * NB_] = run;
        run += c;
    }
    bin_total[(size_t)b * 1024 + h * NB_ + bin] = run;
}

// =====================================================================
// k3b: exclusive scan over the 1024 global bins per batch
// =====================================================================
__global__ void k3b_offsets(const int* __restrict__ bin_total,
                            int* __restrict__ bin_offset) {
    if (threadIdx.x != 0) return;
    const int b = blockIdx.x;
    int run = 0;
    for (int bin = 0; bin < 1024; ++bin) {
        bin_offset[b * 1024 + bin] = run;
        run += bin_total[b * 1024 + bin];
    }
}

// =====================================================================
// k4: stable scatter -> st (sorted positions) and undo_sort output
// =====================================================================
__global__ __launch_bounds__(256)
void k4_scatter(const int* __restrict__ buckets, const int* __restrict__ blockpref,
                const int* __restrict__ bin_offset,
                int* __restrict__ st, float* __restrict__ undo_out) {
    __shared__ int lb[256];
    const int blk = blockIdx.x, h = blockIdx.y, b = blockIdx.z;
    const int t = threadIdx.x;
    const int pos = blk * 256 + t;
    const int bv = buckets[(size_t)b * (NH_ * S_) + (size_t)h * S_ + pos]; // h*128+bucket
    lb[t] = bv;
    __syncthreads();
    int lrank = 0;
    for (int j = 0; j < t; ++j) lrank += (lb[j] == bv);
    const int bin = bv & (NB_ - 1);
    const int p = bin_offset[b * 1024 + bv]
                + blockpref[((((size_t)b * NH_ + h) * NPOSBLK_) + blk) * NB_ + bin]
                + lrank;
    st[(size_t)b * (NH_ * S_) + p] = pos;
    undo_out[(size_t)b * (NH_ * S_) + (size_t)h * S_ + pos] = (float)p;
}

// =====================================================================
// k5: per-(b,chunk) dots = Q·normalize(K)^T * scale, with masks (WMMA)
// grid: (NCH, B) x 256; dyn LDS: Qs 64x128 | Ks 128x128 | qpos|kpos|norms
// Row gathers done by the TDM in gather mode: 16-bit row indices packed
// into D# groups 2/3, 16 rows per descriptor, tile width = 128 elems.
// =====================================================================
__global__ __launch_bounds__(256)
void k5_dots(const float* __restrict__ vecs, const int* __restrict__ st,
             float* __restrict__ dots) {
    extern __shared__ float smem[];
    float* Qs    = smem;                      // [64][128]
    float* Ks    = Qs + 64 * 128;             // [128][128]
    int*   qpos  = (int*)(Ks + 128 * 128);    // [64]
    int*   kpos  = qpos + 64;                 // [128]
    float* norms = (float*)(kpos + 128);      // [128]

    const int c = blockIdx.x, b = blockIdx.y;
    const int t = threadIdx.x;
    const int lane = t & 31, wave = t >> 5;
    const int prev = (c == 0) ? (NCH_ - 1) : (c - 1);
    const int* stb = st + (size_t)b * (NH_ * S_);

    if (t < 64) qpos[t] = stb[c * M_ + t];
    else if (t < 192) {
        const int j = t - 64;
        kpos[j] = (j < 64) ? stb[c * M_ + j] : stb[prev * M_ + (j - 64)];
    }

    if (wave == 0) {
        const float* base = vecs + (size_t)b * S_ * D_;   // row-index gather base
        const unsigned ks_off = (unsigned)(__UINTPTR_TYPE__)(void*)Ks;
        const unsigned qs_off = (unsigned)(__UINTPTR_TYPE__)(void*)Qs;
        const i32x8 g1 = tdm_g1(/*t0=*/D_, /*t1=*/S_, /*tile0=*/D_,
                                /*tile1=*/16, /*stride0=*/D_);
        // K rows: 8 gather descriptors x 16 rows
        #pragma unroll
        for (int d = 0; d < 8; ++d) {
            int idx[16];
            #pragma unroll
            for (int j = 0; j < 16; ++j) {
                const int g = d * 16 + j;
                idx[j] = (g < 64) ? stb[c * M_ + g] : stb[prev * M_ + (g - 64)];
            }
            tdm_issue(tdm_g0(ks_off + d * 16 * D_ * 4, base, 0x80000000u),
                      g1, tdm_pack_idx8(idx), tdm_pack_idx8(idx + 8));
        }
        // Q rows: 4 gather descriptors x 16 rows
        #pragma unroll
        for (int d = 0; d < 4; ++d) {
            int idx[16];
            #pragma unroll
            for (int j = 0; j < 16; ++j)
                idx[j] = stb[c * M_ + d * 16 + j];
            tdm_issue(tdm_g0(qs_off + d * 16 * D_ * 4, base, 0x80000000u),
                      g1, tdm_pack_idx8(idx), tdm_pack_idx8(idx + 8));
        }
        __builtin_amdgcn_s_wait_tensorcnt(0);
    }
    __syncthreads();

    // per-row L2 norms, then in-place normalize (true division, like ref)
    if (t < 128) {
        float s = 0.f;
        const float* row = Ks + t * 128;
        for (int e = 0; e < 128; ++e) { const float v = row[e]; s += v * v; }
        norms[t] = fmaxf(sqrtf(s), 1e-12f);
    }
    __syncthreads();
    for (int idx = t; idx < 128 * 128; idx += 256)
        Ks[idx] = Ks[idx] / norms[idx >> 7];
    __syncthreads();

    // WMMA: wave -> M-tile (wave>>1), 4 N-tiles ((wave&1)*4 ..)
    const int mt  = wave >> 1;
    const int nb0 = (wave & 1) * 4;
    const int lrow = lane & 15;
    const int hi   = lane >> 4;
    v8f acc[4];
    #pragma unroll
    for (int i = 0; i < 4; ++i) acc[i] = (v8f){0.f,0.f,0.f,0.f,0.f,0.f,0.f,0.f};

    const float* qrow = Qs + (mt * 16 + lrow) * 128;
    for (int kk = 0; kk < 32; ++kk) {
        const int k0 = kk * 4 + hi * 2;
        const v2f a = *(const v2f*)(qrow + k0);
        #pragma unroll
        for (int tt = 0; tt < 4; ++tt) {
            // B[k][n] = Ks[key row n][k] : contiguous pair -> v2f LDS load
            const v2f bb = *(const v2f*)(Ks + (size_t)((nb0 + tt) * 16 + lrow) * 128 + k0);
            acc[tt] = wmma_f32_k4(acc[tt], a, bb);
        }
    }

    // epilogue: scale + causal/self masks, write dots[b][c][row][col]
    float* outb = dots + (((size_t)b * NCH_ + c) * M_) * NKEY_;
    #pragma unroll
    for (int tt = 0; tt < 4; ++tt) {
        const int col = (nb0 + tt) * 16 + lrow;
        const int kp = kpos[col];
        #pragma unroll
        for (int r = 0; r < 8; ++r) {
            const int row = mt * 16 + hi * 8 + r;
            const int qp = qpos[row];
            float v = acc[tt][r] * SCALE_;
            if (qp <  kp) v = NEG_FLT_MAX;
            if (qp == kp) v = SELF_ATTN;
            outb[(size_t)row * NKEY_ + col] = v;
        }
    }
}

// =====================================================================
// k6: gather bv (look-one-back) + bq_t + bkv_t     grid: (NCH, B) x 256
// pure streaming copy: direct global->global float4 path is optimal
// =====================================================================
__global__ __launch_bounds__(256)
void k6_bv(const float* __restrict__ vsrc, const int* __restrict__ st,
           float* __restrict__ bv, float* __restrict__ bqt,
           float* __restrict__ bkvt) {
    const int c = blockIdx.x, b = blockIdx.y;
    const int t = threadIdx.x;
    const int prev = (c == 0) ? (NCH_ - 1) : (c - 1);
    const int* stb = st + (size_t)b * (NH_ * S_);

    const int j = t >> 1, c0 = (t & 1) * 64;
    const int pos = (j < 64) ? stb[c * M_ + j] : stb[prev * M_ + (j - 64)];
    const float* src = vsrc + ((size_t)b * S_ + pos) * D_ + c0;
    float* dst = bv + (((size_t)b * NCH_ + c) * NKEY_ + j) * (size_t)D_ + c0;
    #pragma unroll
    for (int q = 0; q < 16; ++q)
        ((float4*)dst)[q] = ((const float4*)src)[q];

    if (t < 64)
        bqt[((size_t)b * NCH_ + c) * M_ + t] = (float)stb[c * M_ + t];
    if (t < 128) {
        const int pj = (t < 64) ? stb[c * M_ + t] : stb[prev * M_ + (t - 64)];
        bkvt[((size_t)b * NCH_ + c) * NKEY_ + t] = (float)pj;
    }
}

// =====================================================================
extern "C" void kernel_launch(void* const* d_in, const int* in_sizes, int n_in,
                              void* d_out, int out_size, void* d_ws, size_t ws_size,
                              hipStream_t stream) {
    const float* vecs = (const float*)d_in[0];   // [8,8192,128]
    const float* v    = (const float*)d_in[1];   // [8,8192,128]
    const float* rot  = (const float*)d_in[2];   // [1,128,8,64]

    // workspace carve (ints)
    int* ws = (int*)d_ws;
    int* buckets   = ws;                      // B*NH*S          = 524288
    int* blockhist = buckets   + 524288;      // B*NH*32*128     = 262144
    int* bin_total = blockhist + 262144;      // B*1024          = 8192
    int* bin_offset= bin_total + 8192;        // B*1024          = 8192
    int* st        = bin_offset+ 8192;        // B*NH*S          = 524288

    // output carve (floats, return order)
    float* out      = (float*)d_out;
    float* dots_o   = out;                                  // 8*1024*64*128
    float* undo_o   = dots_o + (size_t)B_ * NCH_ * M_ * NKEY_;
    float* bqt_o    = undo_o + (size_t)B_ * NH_ * S_;
    float* bkvt_o   = bqt_o  + (size_t)B_ * NCH_ * M_;
    float* bv_o     = bkvt_o + (size_t)B_ * NCH_ * NKEY_;

    const size_t smem1 = (64 * 128 + 128 * 64 + 64 * 64) * sizeof(float);           // 80 KB
    const size_t smem5 = (64 * 128 + 128 * 128) * sizeof(float)
                       + (64 + 128) * sizeof(int) + 128 * sizeof(float);            // ~97 KB

    k1_buckets<<<dim3(S_ / 64, NH_, B_), 256, smem1, stream>>>(vecs, rot, buckets);
    k2_hist   <<<dim3(NPOSBLK_, NH_, B_), 256, 0, stream>>>(buckets, blockhist);
    k3a_scanblocks<<<dim3((B_ * NH_ * NB_ + 255) / 256), 256, 0, stream>>>(blockhist, bin_total);
    k3b_offsets<<<dim3(B_), 64, 0, stream>>>(bin_total, bin_offset);
    k4_scatter<<<dim3(NPOSBLK_, NH_, B_), 256, 0, stream>>>(buckets, blockhist, bin_offset,
                                                            st, undo_o);
    k5_dots   <<<dim3(NCH_, B_), 256, smem5, stream>>>(vecs, st, dots_o);
    k6_bv     <<<dim3(NCH_, B_), 256, 0, stream>>>(v, st, bv_o, bqt_o, bkvt_o);
}